// GNNModel_49503793054393
// MI455X (gfx1250) — compile-verified
//
#include <hip/hip_runtime.h>
#include <hip/hip_bf16.h>

typedef __attribute__((ext_vector_type(2))) float v2f;
typedef __attribute__((ext_vector_type(8))) float v8f;

#define IN_DIM 64

// ---------------- zero init ----------------
__global__ void zero_kernel(float* p, long long count) {
    long long i = (long long)blockIdx.x * blockDim.x + threadIdx.x;
    long long stride = (long long)gridDim.x * blockDim.x;
    for (; i < count; i += stride) p[i] = 0.0f;
}

// ---------------- scatter-add (gather x[src], atomic add into agg[dst]) ----------------
// 32 lanes per edge, each lane owns 2 of the 64 feature dims.
__global__ void scatter_kernel(const float* __restrict__ feat,
                               const long long* __restrict__ src,
                               const long long* __restrict__ dst,
                               float* __restrict__ agg,
                               float* __restrict__ deg,
                               int E, int addDeg) {
    long long gid = (long long)blockIdx.x * blockDim.x + threadIdx.x;
    int e    = (int)(gid >> 5);
    int lane = (int)(gid & 31);
    if (e >= E) return;
    int s = (int)src[e];
    int d = (int)dst[e];
    const float* f = feat + (size_t)s * IN_DIM + lane * 2;
    float*       a = agg  + (size_t)d * IN_DIM + lane * 2;
    atomicAdd(a + 0, f[0]);
    atomicAdd(a + 1, f[1]);
    if (addDeg && lane == 0) atomicAdd(deg + d, 1.0f);
}

// ---------------- fused SAGE layer: out = act( (agg/deg)@Wl + b + A2@Wr ) ----------------
// One wave computes 16 rows x NCOLS cols using V_WMMA_F32_16X16X4_F32.
// A fragment (16x4 f32): lanes 0-15 hold M=0..15 with K={k0,k0+1}; lanes 16-31 with K={k0+2,k0+3}.
// C/D (16x16 f32): vgpr v -> row m0 + v + 8*(lane>=16), col = lane&15.
template <int NCOLS, bool RELU>
__global__ void sage_gemm_kernel(const float* __restrict__ agg,
                                 const float* __restrict__ deg,
                                 const float* __restrict__ A2,
                                 const float* __restrict__ Wl,   // [64 x NCOLS]
                                 const float* __restrict__ bias, // [NCOLS]
                                 const float* __restrict__ Wr,   // [64 x NCOLS]
                                 float* __restrict__ out,        // [N x NCOLS]
                                 int N) {
    extern __shared__ float lds[];          // Wl then Wr, each 64*NCOLS floats
    float* ldsWl = lds;
    float* ldsWr = lds + 64 * NCOLS;

    // cooperative stage of both weight matrices into LDS
    for (int i = threadIdx.x; i < 64 * NCOLS; i += blockDim.x) {
        ldsWl[i] = Wl[i];
        ldsWr[i] = Wr[i];
    }
    __syncthreads();

    const int wave = threadIdx.x >> 5;
    const int lane = threadIdx.x & 31;
    const int wavesPerBlock = blockDim.x >> 5;

    long long rowblock = (long long)blockIdx.x * wavesPerBlock + wave;
    if (rowblock * 16 >= N) return;         // wave-uniform: EXEC stays all-ones

    const int m0   = (int)(rowblock * 16);
    const int hf   = lane >> 4;             // 0: K even pair, 1: K odd pair (+2)
    const int mrow = m0 + (lane & 15);      // row this lane supplies for A
    const int nloc = lane & 15;

    const float dval = deg[mrow];
    const float dinv = 1.0f / fmaxf(dval, 1.0f);

    // Preload both A operands for all 16 K-chunks (K=64)
    v2f a1[16], a2[16];
#pragma unroll
    for (int kc = 0; kc < 16; ++kc) {
        const int k = kc * 4 + hf * 2;
        const float* r1 = agg + (size_t)mrow * 64 + k;
        const float* r2 = A2  + (size_t)mrow * 64 + k;
        a1[kc].x = r1[0] * dinv;
        a1[kc].y = r1[1] * dinv;
        a2[kc].x = r2[0];
        a2[kc].y = r2[1];
    }

#pragma unroll
    for (int nt = 0; nt < NCOLS / 16; ++nt) {
        const int n = nt * 16 + nloc;
        v8f c = {};
        // (agg/deg) @ Wl
#pragma unroll
        for (int kc = 0; kc < 16; ++kc) {
            const int k = kc * 4 + hf * 2;
            v2f b;
            b.x = ldsWl[(k + 0) * NCOLS + n];
            b.y = ldsWl[(k + 1) * NCOLS + n];
            c = __builtin_amdgcn_wmma_f32_16x16x4_f32(
                    false, a1[kc], false, b, (short)0, c, false, false);
        }
        // + A2 @ Wr (accumulate into same C)
#pragma unroll
        for (int kc = 0; kc < 16; ++kc) {
            const int k = kc * 4 + hf * 2;
            v2f b;
            b.x = ldsWr[(k + 0) * NCOLS + n];
            b.y = ldsWr[(k + 1) * NCOLS + n];
            c = __builtin_amdgcn_wmma_f32_16x16x4_f32(
                    false, a2[kc], false, b, (short)0, c, false, false);
        }
        const float bv = bias[n];
#pragma unroll
        for (int v = 0; v < 8; ++v) {
            float o = c[v] + bv;
            if (RELU) o = fmaxf(o, 0.0f);
            const int m = m0 + v + hf * 8;
            if (m < N) out[(size_t)m * NCOLS + n] = o;
        }
    }
}

extern "C" void kernel_launch(void* const* d_in, const int* in_sizes, int n_in,
                              void* d_out, int out_size, void* d_ws, size_t ws_size,
                              hipStream_t stream) {
    const float*     x   = (const float*)d_in[0];
    const long long* ei  = (const long long*)d_in[1];   // int64 [2, E]
    const float*     W1l = (const float*)d_in[2];
    const float*     b1l = (const float*)d_in[3];
    const float*     W1r = (const float*)d_in[4];
    const float*     W2l = (const float*)d_in[5];
    const float*     b2l = (const float*)d_in[6];
    const float*     W2r = (const float*)d_in[7];

    const int N = in_sizes[0] / IN_DIM;
    const int E = in_sizes[1] / 2;
    const long long* src = ei;
    const long long* dst = ei + E;

    // workspace layout: agg [N*64] | deg [N] | h [N*64]
    float* agg = (float*)d_ws;
    float* deg = agg + (size_t)N * 64;
    float* h   = deg + N;
    float* out = (float*)d_out;

    const int ZB = 256;
    long long zcount1 = (long long)N * 64 + N;  // agg + deg contiguous
    int zgrid1 = (int)((zcount1 + ZB - 1) / ZB);
    if (zgrid1 > 65535 * 16) zgrid1 = 65535 * 16;

    long long sthreads = (long long)E * 32;
    int sgrid = (int)((sthreads + 255) / 256);

    const int WAVES = 4;                       // 128 threads per block
    int rowblocks = (N + 15) / 16;
    int ggrid = (rowblocks + WAVES - 1) / WAVES;

    // ---- Layer 1 ----
    zero_kernel<<<zgrid1, ZB, 0, stream>>>(agg, zcount1);
    scatter_kernel<<<sgrid, 256, 0, stream>>>(x, src, dst, agg, deg, E, 1);
    sage_gemm_kernel<64, true><<<ggrid, WAVES * 32, 2 * 64 * 64 * sizeof(float), stream>>>(
        agg, deg, x, W1l, b1l, W1r, h, N);

    // ---- Layer 2 ----
    long long zcount2 = (long long)N * 64;
    int zgrid2 = (int)((zcount2 + ZB - 1) / ZB);
    zero_kernel<<<zgrid2, ZB, 0, stream>>>(agg, zcount2);
    scatter_kernel<<<sgrid, 256, 0, stream>>>(h, src, dst, agg, deg, E, 0);
    sage_gemm_kernel<32, false><<<ggrid, WAVES * 32, 2 * 64 * 32 * sizeof(float), stream>>>(
        agg, deg, h, W2l, b2l, W2r, out, N);
}